// MFVIDependency_67138928771662
// MI455X (gfx1250) — compile-verified
//
#include <hip/hip_runtime.h>
#include <math.h>

// MI455X / gfx1250, wave32.
// Problem: B=8, S=160. Memory-bound iterative contraction over two 131MB f32
// tensors. f2 matvec is computed with V_WMMA_F32_16X16X4_F32 (f32-exact).

#define BB 8
#define SS 160
#define SSS (SS * SS)          // 25600
#define NBD (BB * SS)          // 1280
#define QELEMS (BB * SS * SS)  // 204800

typedef __attribute__((ext_vector_type(2))) float v2f;
typedef __attribute__((ext_vector_type(8))) float v8f;

// ---------------------------------------------------------------- init q0
__global__ __launch_bounds__(256) void k_init_q(float* __restrict__ q,
                                                const unsigned char* __restrict__ mask) {
    int i = blockIdx.x * 256 + threadIdx.x;
    if (i >= QELEMS) return;
    int b = i / SSS;
    int h = (i / SS) % SS;
    bool mh = (h == 0) || (mask[b * SS + h] != 0);  // m.at[:,0] forced True
    q[i] = mh ? 0.0f : -INFINITY;
}

// ------------------------------------------------ softmax over axis h (axis=1)
// One thread per (b,s) column; q[b,h,s] has stride SS in h -> loads are
// perfectly coalesced across adjacent-s lanes. transposed=1 writes the final
// output layout out[b, s, h] (== exp(log_softmax(q^T))).
__global__ __launch_bounds__(256) void k_softmax_h(const float* __restrict__ q,
                                                   float* __restrict__ out,
                                                   int transposed) {
    int g = blockIdx.x * 256 + threadIdx.x;
    if (g >= NBD) return;
    int b = g / SS, s = g % SS;
    const float* col = q + (size_t)b * SSS + s;  // stride SS
    float mx = -INFINITY;
    for (int h = 0; h < SS; ++h) mx = fmaxf(mx, col[(size_t)h * SS]);
    float sum = 0.0f;
    for (int h = 0; h < SS; ++h) sum += __expf(col[(size_t)h * SS] - mx);
    float inv = 1.0f / sum;
    if (!transposed) {
        float* oc = out + (size_t)b * SSS + s;
        for (int h = 0; h < SS; ++h) oc[(size_t)h * SS] = __expf(col[(size_t)h * SS] - mx) * inv;
    } else {
        float* orow = out + (size_t)(b * SS + s) * SS;
        for (int h = 0; h < SS; ++h) orow[h] = __expf(col[(size_t)h * SS] - mx) * inv;
    }
}

// --------------------------------------------------------------- main iteration
// One block per (b,d). 320 threads = 10 wave32s. Wave w owns rows
// h in [16w, 16w+16): f1 row-dots (VALU, shuffle-reduced) and the f2 matvec
// M-tile via V_WMMA_F32_16X16X4_F32.
__global__ __launch_bounds__(320) void k_iter(const float* __restrict__ s_arc,
                                              const float* __restrict__ s_sib,
                                              const float* __restrict__ s_grd,
                                              const unsigned char* __restrict__ mask,
                                              const float* __restrict__ p,
                                              float* __restrict__ q) {
    __shared__ float vsh[SS];   // v_d[s] = p[b,d,s] * (d!=s) * m[s]
    __shared__ float f1sh[SS];
    __shared__ float f2sh[SS];

    int bd = blockIdx.x;        // 0..1279
    int b = bd / SS, d = bd % SS;
    int tid = threadIdx.x;
    int lane = tid & 31, wave = tid >> 5;

    const float* Sblk = s_sib + (size_t)bd * SSS;  // s_sib[b,d,:,:]
    const float* Gblk = s_grd + (size_t)bd * SSS;  // s_grd[b,d,:,:]
    const float* prow_d = p + (size_t)(b * SS + d) * SS;

    if (tid < SS) {
        int s = tid;
        bool ms = (s == 0) || (mask[b * SS + s] != 0);
        vsh[s] = (ms && s != d) ? prow_d[s] : 0.0f;
    }
    __syncthreads();

    // ---- f1[h] = sum_s sib[h,s] * p[b,h,s] * ((h!=s)&(d!=s)&m[s])
    for (int r = 0; r < 16; ++r) {
        int h = wave * 16 + r;
        const float* srow = Sblk + (size_t)h * SS;
        const float* prow = p + (size_t)(b * SS + h) * SS;
        __builtin_prefetch(srow + SS, 0, 1);   // global_prefetch_b8, stream ahead
        float acc = 0.0f;
#pragma unroll
        for (int k = 0; k < 5; ++k) {
            int s = lane + 32 * k;
            bool ms = (s == 0) || (mask[b * SS + s] != 0);
            float w = (ms && s != h && s != d) ? 1.0f : 0.0f;
            acc += srow[s] * prow[s] * w;
        }
#pragma unroll
        for (int off = 16; off; off >>= 1) acc += __shfl_xor(acc, off, 32);
        if (lane == 0) f1sh[h] = acc;
    }

    // ---- f2 full matvec via WMMA f32 16x16x4: D(:,0) = G_tile @ v
    {
        int m0 = wave * 16;
        int m = lane & 15;
        int half = lane >> 4;                 // 0: K{0,1}, 1: K{2,3}
        const float* arow = Gblk + (size_t)(m0 + m) * SS;
        v8f acc = {};
        for (int k0 = 0; k0 < SS; k0 += 4) {
            // A 16x4 f32: lanes 0-15 hold (K0,K1), lanes 16-31 hold (K2,K3)
            v2f a = *(const v2f*)(arow + k0 + (half << 1));
            // B 4x16 f32: VGPR0 row K = half, VGPR1 row K = 2 + half; only N=0 live
            float b0 = vsh[k0 + half];
            float b1 = vsh[k0 + 2 + half];
            v2f bm;
            bm.x = (m == 0) ? b0 : 0.0f;
            bm.y = (m == 0) ? b1 : 0.0f;
            acc = __builtin_amdgcn_wmma_f32_16x16x4_f32(
                /*neg_a=*/false, a, /*neg_b=*/false, bm,
                /*c_mod=*/(short)0, acc, /*reuse_a=*/false, /*reuse_b=*/false);
        }
        // D column N=0 lives in lanes 0 (M = m0+g) and 16 (M = m0+8+g)
        if ((lane & 15) == 0) {
            int base = m0 + (lane >> 4) * 8;
#pragma unroll
            for (int g = 0; g < 8; ++g) f2sh[base + g] = acc[g];
        }
    }
    __syncthreads();

    // ---- q_new[b,h,d] = mh ? arc[b,h,d] + md*(f1 + f2 - G[h,h]*v[h]) : -inf
    if (tid < SS) {
        int h = tid;
        bool mh = (h == 0) || (mask[b * SS + h] != 0);
        bool md = (d == 0) || (mask[b * SS + d] != 0);
        float corr = Gblk[(size_t)h * SS + h] * vsh[h];  // removes s==h term
        float f = md ? (f1sh[h] + f2sh[h] - corr) : 0.0f;
        float arc = s_arc[(size_t)bd * SS + h];           // s_arc[b,d,h]
        q[(size_t)(b * SS + h) * SS + d] = mh ? (arc + f) : -INFINITY;
    }
}

// ----------------------------------------------------------------- launch
extern "C" void kernel_launch(void* const* d_in, const int* in_sizes, int n_in,
                              void* d_out, int out_size, void* d_ws, size_t ws_size,
                              hipStream_t stream) {
    const float* s_arc = (const float*)d_in[0];
    const float* s_sib = (const float*)d_in[1];
    const float* s_grd = (const float*)d_in[2];
    const unsigned char* mask = (const unsigned char*)d_in[3];  // numpy bool_ = 1 byte
    float* out = (float*)d_out;

    float* q = (float*)d_ws;          // QELEMS floats
    float* p = q + QELEMS;            // QELEMS floats (needs 1.6 MB total)

    k_init_q<<<(QELEMS + 255) / 256, 256, 0, stream>>>(q, mask);
    for (int it = 0; it < 3; ++it) {
        k_softmax_h<<<NBD / 256, 256, 0, stream>>>(q, p, 0);
        k_iter<<<NBD, 320, 0, stream>>>(s_arc, s_sib, s_grd, mask, p, q);
    }
    // final: exp(log_softmax(q^T)) == transposed column softmax
    k_softmax_h<<<NBD / 256, 256, 0, stream>>>(q, out, 1);
}